// MultiScaleMixtureGLR_69148973465847
// MI455X (gfx1250) — compile-verified
//
#include <hip/hip_runtime.h>
#include <math.h>

typedef __attribute__((ext_vector_type(2))) float v2f;
typedef __attribute__((ext_vector_type(8))) float v8f;

constexpr int B = 4, G = 4, F = 8, C = 3, H = 256, W = 256;
constexpr int E = 9, C3 = 9;
constexpr int HW = H * W;
constexpr int BG = B * G;

static __device__ __forceinline__ int clampi(int v, int lo, int hi) {
  return v < lo ? lo : (v > hi ? hi : v);
}
// reflect pad=1 (jnp 'reflect'): -1 -> 1, N -> N-2
static __device__ __forceinline__ int reflect1(int i, int n) {
  return i < 0 ? -i : (i >= n ? 2 * n - 2 - i : i);
}

// ---------------- Pass 1: xt = (x / max(||x||_F, 1e-12)) * multiM -----------
__global__ void k_xt(const float* __restrict__ xin,
                     const float* __restrict__ multiM,
                     float* __restrict__ xt) {
  int idx = blockIdx.x * blockDim.x + threadIdx.x;
  if (idx >= BG * HW) return;
  int hw = idx % HW;
  int bg = idx / HW;
  int g = bg % G;
  const float* src = xin + (size_t)bg * F * HW + hw;
  float v[F];
  float ss = 0.f;
#pragma unroll
  for (int f = 0; f < F; ++f) { v[f] = src[(size_t)f * HW]; ss += v[f] * v[f]; }
  float inv = 1.0f / fmaxf(sqrtf(ss), 1e-12f);
  float* dst = xt + (size_t)bg * F * HW + hw;
#pragma unroll
  for (int f = 0; f < F; ++f) dst[(size_t)f * HW] = v[f] * inv * multiM[g * F + f];
}

// ---------------- Pass 2: edge weights ew[9] and nf = 1/sqrt(deg) -----------
__global__ void k_edge(const float* __restrict__ xt,
                       float* __restrict__ ew,
                       float* __restrict__ nf) {
  int idx = blockIdx.x * blockDim.x + threadIdx.x;
  if (idx >= BG * HW) return;
  int w = idx % W;
  int h = (idx / W) % H;
  int bg = idx / HW;
  const float* base = xt + (size_t)bg * F * HW;
  float c[F];
#pragma unroll
  for (int f = 0; f < F; ++f) c[f] = base[(size_t)f * HW + h * W + w];
  float deg = 0.f;
  float ewv[E];
#pragma unroll
  for (int e = 0; e < E; ++e) {
    int hh = clampi(h + e / 3 - 1, 0, H - 1);   // replicate (edge) pad
    int ww = clampi(w + e % 3 - 1, 0, W - 1);
    float s = 0.f;
#pragma unroll
    for (int f = 0; f < F; ++f) s += c[f] * base[(size_t)f * HW + hh * W + ww];
    s = fminf(fmaxf(s, -10.f), 10.f);
    float ev = expf(s);
    ewv[e] = ev;
    deg += ev;
  }
  float* ewo = ew + (size_t)bg * E * HW + h * W + w;
#pragma unroll
  for (int e = 0; e < E; ++e) ewo[(size_t)e * HW] = ewv[e];
  nf[(size_t)bg * HW + h * W + w] = 1.0f / sqrtf(deg);
}

// ---------------- Pass 3: grouped stats conv via V_WMMA_F32_16X16X4_F32 -----
// sig[9 x 16px] = A[9x27 block-diagonal stats kernel] x col[27x16 im2col taps]
// One wave per 16-pixel strip; K=27 padded to 28 -> 7 chained WMMAs.
__global__ void k_sig_wmma(const float* __restrict__ patchs,
                           const float* __restrict__ sk,   // [9][9] = [3C,1,3,3]
                           float* __restrict__ sig) {
  const int lane = threadIdx.x & 31;
  const int wave = threadIdx.x >> 5;
  const int half = lane >> 4;   // K sub-pair selector (ISA 16x4 f32 A layout)
  const int lp   = lane & 15;   // M row (A) / N column (B,C,D)
  const int tile = blockIdx.x * 8 + wave;  // 0..15 strips per row
  const int h    = blockIdx.y;
  const int bg   = blockIdx.z;
  const int w0   = tile * 16;
  const float* p = patchs + (size_t)bg * C * HW;

  v8f acc = {0.f, 0.f, 0.f, 0.f, 0.f, 0.f, 0.f, 0.f};
#pragma unroll
  for (int s = 0; s < 7; ++s) {
    const int k0 = 4 * s + 2 * half;
    float av[2], bv[2];
#pragma unroll
    for (int j = 0; j < 2; ++j) {
      const int k = k0 + j;
      // A[m,k]: nonzero only for m<9, k<27, and k's channel == m's group
      float a = 0.f;
      if (lp < 9 && k < 27 && (k / 9) == (lp / 3)) a = sk[lp * 9 + (k % 9)];
      av[j] = a;
      // col[k, n=lp]: reflect-padded patch tap
      float bb = 0.f;
      if (k < 27) {
        const int cch = k / 9, t = k % 9;
        const int hr = reflect1(h + t / 3 - 1, H);
        const int wr = reflect1(w0 + lp + t % 3 - 1, W);
        bb = p[(size_t)cch * HW + hr * W + wr];
      }
      bv[j] = bb;
    }
    v2f a = {av[0], av[1]};
    v2f b = {bv[0], bv[1]};
    acc = __builtin_amdgcn_wmma_f32_16x16x4_f32(
        /*neg_a=*/false, a, /*neg_b=*/false, b,
        /*c_mod=*/(short)0, acc, /*reuse_a=*/false, /*reuse_b=*/false);
  }
  // D layout: lanes 0-15 hold M=0..7 (v0..v7) for N=lp; lanes 16-31 hold M=8..15
  float* so = sig + (size_t)bg * C3 * HW + h * W + (w0 + lp);
  if (half == 0) {
#pragma unroll
    for (int i = 0; i < 8; ++i) so[(size_t)i * HW] = acc[i];
  } else {
    so[(size_t)8 * HW] = acc[0];  // channel 8; rows 9..15 are padding
  }
}

// ---------------- Pass 4: out = sig - sum_e ewn[e] * sig_nb[e] --------------
__global__ void k_glr(const float* __restrict__ sig,
                      const float* __restrict__ ew,
                      const float* __restrict__ nf,
                      float* __restrict__ outb) {
  int idx = blockIdx.x * blockDim.x + threadIdx.x;
  if (idx >= BG * HW) return;
  int w = idx % W;
  int h = (idx / W) % H;
  int bg = idx / HW;
  int hw = h * W + w;
  const float* sg  = sig + (size_t)bg * C3 * HW;
  const float* e0  = ew + (size_t)bg * E * HW;
  const float* nfp = nf + (size_t)bg * HW;
  float acc[C3];
#pragma unroll
  for (int c = 0; c < C3; ++c) acc[c] = sg[(size_t)c * HW + hw];
  float nfc = nfp[hw];
#pragma unroll
  for (int e = 0; e < E; ++e) {
    int hh = clampi(h + e / 3 - 1, 0, H - 1);
    int ww = clampi(w + e % 3 - 1, 0, W - 1);
    int hwn = hh * W + ww;
    float wgt = nfc * e0[(size_t)e * HW + hw] * nfp[hwn];
#pragma unroll
    for (int c = 0; c < C3; ++c) acc[c] -= wgt * sg[(size_t)c * HW + hwn];
  }
  float* o = outb + (size_t)bg * C3 * HW + hw;
#pragma unroll
  for (int c = 0; c < C3; ++c) o[(size_t)c * HW] = acc[c];
}

// ---------------- Pass 5: grouped transpose-conv (flipped kernel, zero pad) -
__global__ void k_tconv(const float* __restrict__ outb,
                        const float* __restrict__ sk,
                        float* __restrict__ y) {
  int idx = blockIdx.x * blockDim.x + threadIdx.x;
  if (idx >= BG * HW) return;
  int w = idx % W;
  int h = (idx / W) % H;
  int bg = idx / HW;
  const float* ob = outb + (size_t)bg * C3 * HW;
#pragma unroll
  for (int c = 0; c < C; ++c) {
    float acc = 0.f;
#pragma unroll
    for (int j = 0; j < 3; ++j) {
      const int ch = 3 * c + j;
#pragma unroll
      for (int t = 0; t < 9; ++t) {
        const int ky = t / 3, kx = t % 3;
        const int hh = h + ky - 1, ww = w + kx - 1;
        if (hh >= 0 && hh < H && ww >= 0 && ww < W) {
          acc += sk[ch * 9 + (2 - ky) * 3 + (2 - kx)] *
                 ob[(size_t)ch * HW + hh * W + ww];
        }
      }
    }
    y[((size_t)bg * C + c) * HW + h * W + w] = acc;
  }
}

extern "C" void kernel_launch(void* const* d_in, const int* in_sizes, int n_in,
                              void* d_out, int out_size, void* d_ws, size_t ws_size,
                              hipStream_t stream) {
  (void)in_sizes; (void)n_in; (void)out_size; (void)ws_size;
  const float* img    = (const float*)d_in[0];
  const float* patchs = (const float*)d_in[1];
  const float* multiM = (const float*)d_in[2];
  const float* sk     = (const float*)d_in[3];

  // Workspace layout (floats). Region A is reused: xt (8.4M) then out (9.4M).
  float* ws   = (float*)d_ws;
  float* xtb  = ws;                                   // region A
  float* ewb  = ws + (size_t)BG * C3 * HW;            // 9,437,184 floats in
  float* nfb  = ewb + (size_t)BG * E * HW;
  float* sigb = nfb + (size_t)BG * HW;
  float* outb = xtb;                                  // reuse region A
  float* yb   = (float*)d_out;

  const int n = BG * HW;
  const int blocks = (n + 255) / 256;

  k_xt<<<blocks, 256, 0, stream>>>(img, multiM, xtb);
  k_edge<<<blocks, 256, 0, stream>>>(xtb, ewb, nfb);
  dim3 g3(2, H, BG);  // 8 waves/block * 2 blocks = 16 strips of 16 px per row
  k_sig_wmma<<<g3, 256, 0, stream>>>(patchs, sk, sigb);
  k_glr<<<blocks, 256, 0, stream>>>(sigb, ewb, nfb, outb);
  k_tconv<<<blocks, 256, 0, stream>>>(outb, sk, yb);
}